// NeRFs_hybrid_64312840290451
// MI455X (gfx1250) — compile-verified
//
#include <hip/hip_runtime.h>
#include <hip/hip_bf16.h>
#include <stdint.h>

// NeRF SDF->weights volume rendering, MI455X (gfx1250, wave32).
// Memory-bound streaming kernel (~200 MB traffic -> ~8.6us floor @ 23.3 TB/s).
// One wave per ray; double-buffered GLOBAL_LOAD_ASYNC_TO_LDS_B128 pipeline
// (ASYNCcnt) hides HBM latency; wave32 shuffle reductions for the S=128 sums.

#define N_SAMPLES        128
#define TRUNC            0.05f
#define WAVES_PER_BLOCK  8
#define BLOCK_SIZE       256
#define RAYS_PER_WAVE    8

// Low 32 bits of a generic pointer to LDS are the LDS byte offset
// (ISA 10.2: flat->LDS mapping uses addr[31:0]).
__device__ __forceinline__ unsigned lds_off(const void* p) {
  return (unsigned)(uintptr_t)p;
}

// GVS-mode async copy: lds[ldsByteOff] <= *(saddr + voffset), 16B per lane.
__device__ __forceinline__ void async_b128(unsigned ldsByteOff, unsigned gByteOff,
                                           const void* base) {
  asm volatile("global_load_async_to_lds_b128 %0, %1, %2"
               :: "v"(ldsByteOff), "v"(gByteOff),
                  "s"((unsigned long long)(uintptr_t)base)
               : "memory");
}

// Issue the 5 async b128 copies (2048B raw row + 512B z row) for one ray.
__device__ __forceinline__ void issue_ray(const float* raw, const float* zv, int ray,
                                          unsigned lRaw, unsigned lZ, int lane) {
  const unsigned l16 = (unsigned)lane * 16u;
  const unsigned gb  = (unsigned)ray * (N_SAMPLES * 4u * 4u);  // 2048 B / ray
#pragma unroll
  for (unsigned j = 0; j < 4; ++j)
    async_b128(lRaw + j * 512u + l16, gb + j * 512u + l16, raw);
  async_b128(lZ + l16, (unsigned)ray * (N_SAMPLES * 4u) + l16, zv);
}

__device__ __forceinline__ float wave_sum(float v) {
#pragma unroll
  for (int off = 16; off > 0; off >>= 1) v += __shfl_xor(v, off, 32);
  return v;
}
__device__ __forceinline__ int wave_min(int v) {
#pragma unroll
  for (int off = 16; off > 0; off >>= 1) v = min(v, __shfl_xor(v, off, 32));
  return v;
}

__device__ __forceinline__ float fsigmoid(float x) {
  return 1.0f / (1.0f + __expf(-x));
}
// sigmoid(s/T)*sigmoid(-s/T) = e / (1+e)^2 with e = exp(-s/T): one exp.
__device__ __forceinline__ float sdf_bump(float s, float invT) {
  float e = __expf(-s * invT);
  float d = 1.0f + e;
  return e / (d * d);
}

__global__ void __launch_bounds__(BLOCK_SIZE)
nerf_sdf_render(const float* __restrict__ raw, const float* __restrict__ zv,
                float* __restrict__ out, int n_rays, int total_waves) {
  __shared__ __align__(16) float s_raw[2][WAVES_PER_BLOCK][N_SAMPLES * 4];  // 32 KB
  __shared__ __align__(16) float s_z[2][WAVES_PER_BLOCK][N_SAMPLES];        //  8 KB

  const int lane = threadIdx.x & 31;
  const int wv   = threadIdx.x >> 5;
  const int wid  = blockIdx.x * WAVES_PER_BLOCK + wv;

  const size_t N = (size_t)n_rays;
  float* rgb_map  = out;                          // [N,3]
  float* disp_map = out + 3 * N;                  // [N]
  float* acc_map  = out + 4 * N;                  // [N]
  float* wgt      = out + 5 * N;                  // [N,128]
  float* depth_o  = out + 5 * N + N * N_SAMPLES;  // [N]
  float* var_o    = out + 6 * N + N * N_SAMPLES;  // [N]

  unsigned ldsRaw[2], ldsZ[2];
#pragma unroll
  for (int b = 0; b < 2; ++b) {
    ldsRaw[b] = lds_off(&s_raw[b][wv][0]);
    ldsZ[b]   = lds_off(&s_z[b][wv][0]);
  }

  const float invT = 1.0f / TRUNC;
  const int   BIG  = 1 << 30;

  int ray = wid;
  if (ray < n_rays) issue_ray(raw, zv, ray, ldsRaw[0], ldsZ[0], lane);

  int buf = 0;
  for (; ray < n_rays; ray += total_waves) {
    const int nxt = ray + total_waves;
    if (nxt < n_rays) {
      issue_ray(raw, zv, nxt, ldsRaw[buf ^ 1], ldsZ[buf ^ 1], lane);
      // 10 outstanding; oldest 5 (current buffer) complete in order.
      asm volatile("s_wait_asynccnt 0x5" ::: "memory");
    } else {
      asm volatile("s_wait_asynccnt 0x0" ::: "memory");
    }

    const float4* pr = (const float4*)&s_raw[buf][wv][0];
    const float4* pz = (const float4*)&s_z[buf][wv][0];
    const float4 z4 = pz[lane];
    const float4 r0 = pr[lane * 4 + 0];
    const float4 r1 = pr[lane * 4 + 1];
    const float4 r2 = pr[lane * 4 + 2];
    const float4 r3 = pr[lane * 4 + 3];

    const float sdf0 = r0.w, sdf1 = r1.w, sdf2 = r2.w, sdf3 = r3.w;

    // First zero-crossing index (argmax of sign mask; all-zero -> 0).
    const float nbr = __shfl_down(sdf0, 1, 32);  // lane l+1's sdf[4(l+1)]
    const int base = lane * 4;
    int cand = BIG;
    if (lane < 31 && sdf3 * nbr < 0.0f) cand = base + 3;
    if (sdf2 * sdf3 < 0.0f) cand = base + 2;
    if (sdf1 * sdf2 < 0.0f) cand = base + 1;
    if (sdf0 * sdf1 < 0.0f) cand = base + 0;
    int m = wave_min(cand);
    const int ind = (m == BIG) ? 0 : m;

    const int owner = ind >> 2, sub = ind & 3;  // sub uniform across wave
    const float zs   = (sub == 0) ? z4.x : (sub == 1) ? z4.y : (sub == 2) ? z4.z : z4.w;
    const float zlim = __shfl(zs, owner, 32) + TRUNC;

    const float w0 = (z4.x < zlim) ? sdf_bump(sdf0, invT) : 0.0f;
    const float w1 = (z4.y < zlim) ? sdf_bump(sdf1, invT) : 0.0f;
    const float w2 = (z4.z < zlim) ? sdf_bump(sdf2, invT) : 0.0f;
    const float w3 = (z4.w < zlim) ? sdf_bump(sdf3, invT) : 0.0f;

    float sw   = w0 + w1 + w2 + w3;
    float swz  = w0 * z4.x + w1 * z4.y + w2 * z4.z + w3 * z4.w;
    float swz2 = w0 * z4.x * z4.x + w1 * z4.y * z4.y + w2 * z4.z * z4.z + w3 * z4.w * z4.w;
    float swr  = w0 * fsigmoid(r0.x) + w1 * fsigmoid(r1.x) + w2 * fsigmoid(r2.x) + w3 * fsigmoid(r3.x);
    float swg  = w0 * fsigmoid(r0.y) + w1 * fsigmoid(r1.y) + w2 * fsigmoid(r2.y) + w3 * fsigmoid(r3.y);
    float swb  = w0 * fsigmoid(r0.z) + w1 * fsigmoid(r1.z) + w2 * fsigmoid(r2.z) + w3 * fsigmoid(r3.z);

    sw   = wave_sum(sw);
    swz  = wave_sum(swz);
    swz2 = wave_sum(swz2);
    swr  = wave_sum(swr);
    swg  = wave_sum(swg);
    swb  = wave_sum(swb);

    const float inv = 1.0f / (sw + 1e-8f);

    float4 wn;
    wn.x = w0 * inv; wn.y = w1 * inv; wn.z = w2 * inv; wn.w = w3 * inv;
    ((float4*)(wgt + (size_t)ray * N_SAMPLES))[lane] = wn;

    if (lane == 0) {
      const float depth = swz * inv;
      const float acc   = sw * inv;
      const float var   = (swz2 - 2.0f * depth * swz + depth * depth * sw) * inv;
      const float disp  = 1.0f / fmaxf(1e-10f, depth / acc);
      rgb_map[3 * (size_t)ray + 0] = swr * inv;
      rgb_map[3 * (size_t)ray + 1] = swg * inv;
      rgb_map[3 * (size_t)ray + 2] = swb * inv;
      disp_map[ray] = disp;
      acc_map[ray]  = acc;
      depth_o[ray]  = depth;
      var_o[ray]    = var;
    }
    buf ^= 1;
  }
}

extern "C" void kernel_launch(void* const* d_in, const int* in_sizes, int n_in,
                              void* d_out, int out_size, void* d_ws, size_t ws_size,
                              hipStream_t stream) {
  const float* raw = (const float*)d_in[0];  // [N,128,4]
  const float* zv  = (const float*)d_in[1];  // [N,128]
  float* out = (float*)d_out;

  const int n_rays = in_sizes[1] / N_SAMPLES;                 // 65536
  int waves_wanted = (n_rays + RAYS_PER_WAVE - 1) / RAYS_PER_WAVE;
  int blocks = (waves_wanted + WAVES_PER_BLOCK - 1) / WAVES_PER_BLOCK;  // 1024
  if (blocks < 1) blocks = 1;
  const int total_waves = blocks * WAVES_PER_BLOCK;           // ray stride

  nerf_sdf_render<<<blocks, BLOCK_SIZE, 0, stream>>>(raw, zv, out, n_rays, total_waves);
}